// ManualRNN_64106681860393
// MI455X (gfx1250) — compile-verified
//
#include <hip/hip_runtime.h>
#include <hip/hip_bf16.h>

// ManualRNN on gfx1250:
//   Phase 1: x_pre[m,n] = embed[x[m]] @ W_x^T + b   (M=B*S=131072, N=K=256), fp32 WMMA
//   Phase 2: per-batch-row sequential scan h = tanh(x_pre + h @ W_h^T),
//            W_h^T resident in LDS (256 KB), double-buffered h-state in LDS,
//            in-place on d_out, one barrier per timestep.

#define HID    256
#define SEQL   2048
#define BATCHN 64
#define LDSPAD 260   // padded row stride (floats) to avoid LDS bank conflicts
#define HBUF   (16 * LDSPAD)

typedef __attribute__((ext_vector_type(2))) float v2f;
typedef __attribute__((ext_vector_type(8))) float v8f;

__device__ __forceinline__ v8f wmma_f32_16x16x4(v2f a, v2f b, v8f c) {
  // 8 args: (neg_a, A, neg_b, B, c_mod, C, reuse_a, reuse_b)
  return __builtin_amdgcn_wmma_f32_16x16x4_f32(false, a, false, b, (short)0, c,
                                               false, false);
}

// ---------------- Phase 1: embedding gather + input projection ----------------
// One block = 256 threads = 8 waves handles one 16-row M-tile across all N=256.
// 16 embedding rows are staged coalesced into LDS; each wave computes two
// 16x16 output tiles (n0 = wave*32, n0+16) with a K=256 fp32-WMMA loop.
__global__ __launch_bounds__(256) void rnn_xproj_kernel(
    const int* __restrict__ x, const float* __restrict__ embed,
    const float* __restrict__ Wx, const float* __restrict__ bx,
    float* __restrict__ xpre) {
  __shared__ float sInp[16 * LDSPAD];

  const int tid   = threadIdx.x;
  const int mBase = blockIdx.x * 16;

  // Stage 16 embedding rows (coalesced: consecutive tids -> consecutive cols).
  for (int idx = tid; idx < 16 * HID; idx += 256) {
    const int r   = idx >> 8;
    const int c   = idx & (HID - 1);
    const int tok = x[mBase + r];
    sInp[r * LDSPAD + c] = embed[(size_t)tok * HID + c];
  }
  __syncthreads();

  const int wave = tid >> 5;
  const int lane = tid & 31;
  const int ln   = lane & 15;   // matrix row/col within tile
  const int half = lane >> 4;   // lane half selects K sub-pair / M+8 rows
  const int n0   = wave * 32;
  const int n1   = n0 + 16;

  v8f acc0, acc1;
  const float b0 = bx[n0 + ln];
  const float b1 = bx[n1 + ln];
#pragma unroll
  for (int r = 0; r < 8; ++r) { acc0[r] = b0; acc1[r] = b1; }

#pragma unroll 4
  for (int k = 0; k < HID; k += 4) {
    const int k0 = k + 2 * half;
    v2f a, w0, w1;
    a.x  = sInp[ln * LDSPAD + k0];
    a.y  = sInp[ln * LDSPAD + k0 + 1];
    // B[k][n] = W_x[n][k]  (row-major W_x, contiguous in k per lane)
    w0.x = Wx[(size_t)(n0 + ln) * HID + k0];
    w0.y = Wx[(size_t)(n0 + ln) * HID + k0 + 1];
    w1.x = Wx[(size_t)(n1 + ln) * HID + k0];
    w1.y = Wx[(size_t)(n1 + ln) * HID + k0 + 1];
    acc0 = wmma_f32_16x16x4(a, w0, acc0);
    acc1 = wmma_f32_16x16x4(a, w1, acc1);
  }

#pragma unroll
  for (int r = 0; r < 8; ++r) {
    const int mrow = mBase + r + 8 * half;
    xpre[(size_t)mrow * HID + n0 + ln] = acc0[r];
    xpre[(size_t)mrow * HID + n1 + ln] = acc1[r];
  }
}

// ---------------- Phase 2: recurrent scan, in-place on d_out ----------------
// 4 blocks x 512 threads (16 waves). Block owns 16 batch rows (independent
// recurrences). LDS: W_h^T as [k][n] (256 KB) + double-buffered h [2][16][PAD].
// Wave w computes output columns n0 = w*16 .. n0+15 each step.
// acc starts at 0 so the x_pre global loads overlap the 64-WMMA LDS loop;
// two independent K-chains (0..127, 128..255) per wave for XDL latency hiding.
__global__ __launch_bounds__(512) void rnn_scan_kernel(
    float* __restrict__ io, const float* __restrict__ Wh) {
  extern __shared__ float smem[];
  float* wl = smem;              // wl[k*HID + n] = Wh[n][k]
  float* hb = smem + HID * HID;  // two h buffers, each 16*LDSPAD

  const int tid = threadIdx.x;

  // Load W_h transposed into LDS (coalesced global reads).
  for (int idx = tid; idx < HID * HID; idx += 512) {
    const int n = idx >> 8;
    const int k = idx & (HID - 1);
    wl[k * HID + n] = Wh[idx];
  }
  // h0 = 0 (both buffers)
  for (int idx = tid; idx < 2 * HBUF; idx += 512) hb[idx] = 0.0f;
  __syncthreads();

  const int wave  = tid >> 5;   // 0..15 -> output column tile
  const int lane  = tid & 31;
  const int ln    = lane & 15;
  const int half  = lane >> 4;
  const int n0    = wave * 16;
  const int bBase = blockIdx.x * 16;

  int cur = 0;
  for (int t = 0; t < SEQL; ++t) {
    // Issue x_pre loads early; they complete while the WMMA loop runs.
    float c[8];
#pragma unroll
    for (int r = 0; r < 8; ++r) {
      const int bRow = bBase + r + 8 * half;
      c[r] = io[((size_t)bRow * SEQL + t) * HID + n0 + ln];
    }

    const float* hl = hb + cur * HBUF;
    v8f acc0, acc1;
#pragma unroll
    for (int r = 0; r < 8; ++r) { acc0[r] = 0.0f; acc1[r] = 0.0f; }

    // h_{t-1} @ W_h^T over K=256, split into two independent chains.
#pragma unroll 4
    for (int k = 0; k < HID / 2; k += 4) {
      const int k0 = k + 2 * half;          // chain 0: K in [0,128)
      const int k1 = k0 + HID / 2;          // chain 1: K in [128,256)
      v2f a0, b0, a1, b1;
      a0.x = hl[ln * LDSPAD + k0];
      a0.y = hl[ln * LDSPAD + k0 + 1];
      a1.x = hl[ln * LDSPAD + k1];
      a1.y = hl[ln * LDSPAD + k1 + 1];
      b0.x = wl[k0 * HID + n0 + ln];
      b0.y = wl[(k0 + 1) * HID + n0 + ln];
      b1.x = wl[k1 * HID + n0 + ln];
      b1.y = wl[(k1 + 1) * HID + n0 + ln];
      acc0 = wmma_f32_16x16x4(a0, b0, acc0);
      acc1 = wmma_f32_16x16x4(a1, b1, acc1);
    }

    float d[8];
#pragma unroll
    for (int r = 0; r < 8; ++r) d[r] = tanhf(acc0[r] + acc1[r] + c[r]);

    // Store state to global (overwrites x_pre slot for this t).
#pragma unroll
    for (int r = 0; r < 8; ++r) {
      const int bRow = bBase + r + 8 * half;
      io[((size_t)bRow * SEQL + t) * HID + n0 + ln] = d[r];
    }

    // Write new h into the *other* buffer: no race with this step's reads.
    float* hn = hb + (cur ^ 1) * HBUF;
#pragma unroll
    for (int r = 0; r < 8; ++r)
      hn[(r + 8 * half) * LDSPAD + n0 + ln] = d[r];

    __syncthreads();   // single barrier: next step may now read hn
    cur ^= 1;
  }
}

extern "C" void kernel_launch(void* const* d_in, const int* in_sizes, int n_in,
                              void* d_out, int out_size, void* d_ws, size_t ws_size,
                              hipStream_t stream) {
  (void)in_sizes; (void)n_in; (void)out_size; (void)d_ws; (void)ws_size;
  const int*   x     = (const int*)d_in[0];    // [B, S] int32
  const float* embed = (const float*)d_in[1];  // [VOCAB, H]
  const float* Wx    = (const float*)d_in[2];  // [H, H]
  const float* bx    = (const float*)d_in[3];  // [H]
  const float* Wh    = (const float*)d_in[4];  // [H, H]
  float* out = (float*)d_out;                  // [B, S, H]

  // Phase 1: 131072 rows / 16 per block
  rnn_xproj_kernel<<<dim3((BATCHN * SEQL) / 16), dim3(256), 0, stream>>>(
      x, embed, Wx, bx, out);

  // Phase 2: 4 blocks of 16 batch rows; LDS = W_h^T (256KB) + 2x h state
  const size_t shmem = (size_t)(HID * HID + 2 * HBUF) * sizeof(float);
  rnn_scan_kernel<<<dim3(BATCHN / 16), dim3(512), shmem, stream>>>(out, Wh);
}